// GIN_21002390077837
// MI455X (gfx1250) — compile-verified
//
#include <hip/hip_runtime.h>
#include <hip/hip_bf16.h>

// ---------------- problem constants (fixed by reference) ----------------
#define DH    128          // hidden dim
#define NL    3            // layers
#define NG    512          // graphs
#define NC    10           // classes
#define DCAT  (DH * NL)    // 384
#define TM    16           // node rows per tile
#define PADH  (DH + 4)     // LDS row stride (conflict-free for stride-4k frag reads)

#ifndef __has_builtin
#define __has_builtin(x) 0
#endif

#if defined(__HIP_DEVICE_COMPILE__) && __has_builtin(__builtin_amdgcn_global_load_async_to_lds_b128)
#define HAVE_ASYNC_TO_LDS 1
#else
#define HAVE_ASYNC_TO_LDS 0
#endif

typedef __attribute__((ext_vector_type(2))) float v2f;
typedef __attribute__((ext_vector_type(4))) float v4f;
typedef __attribute__((ext_vector_type(8))) float v8f;
typedef __attribute__((ext_vector_type(4))) int   v4i;

#if HAVE_ASYNC_TO_LDS
typedef __attribute__((address_space(1))) v4i gas_v4i;   // global (AS1) int4
typedef __attribute__((address_space(3))) v4i las_v4i;   // LDS (AS3) int4
#endif

__device__ __forceinline__ void atomAddF(float* p, float v) {
    // lowers to global_atomic_add_f32 (relaxed, device scope)
    __hip_atomic_fetch_add(p, v, __ATOMIC_RELAXED, __HIP_MEMORY_SCOPE_AGENT);
}

__device__ __forceinline__ void waitAsync0() {
#if defined(__HIP_DEVICE_COMPILE__)
#if __has_builtin(__builtin_amdgcn_s_wait_asynccnt)
    __builtin_amdgcn_s_wait_asynccnt(0);
#else
    asm volatile("s_wait_asynccnt 0x0" ::: "memory");
#endif
#endif
}

// ---------------- utility kernels ----------------
__global__ void gin_zero_kernel(float* __restrict__ p, int n) {
    int i = blockIdx.x * blockDim.x + threadIdx.x;
    if (i < n) p[i] = 0.0f;
}

__global__ void gin_copy4_kernel(float* __restrict__ dst, const float* __restrict__ src, int n4) {
    int i = blockIdx.x * blockDim.x + threadIdx.x;
    if (i < n4) ((v4f*)dst)[i] = ((const v4f*)src)[i];
}

// ---------------- edge scatter: aggr[dst] += h[src] ----------------
// one wave (32 lanes) per edge; each lane owns 4 consecutive floats (float4)
__global__ __launch_bounds__(256)
void gin_edge_kernel(const float* __restrict__ h, const int* __restrict__ srcI,
                     const int* __restrict__ dstI, float* __restrict__ aggr, int nEdges) {
    long long t = (long long)blockIdx.x * blockDim.x + threadIdx.x;
    int e    = (int)(t >> 5);
    int lane = (int)(t & 31);
    if (e >= nEdges) return;
    int s = srcI[e];
    int d = dstI[e];
    const v4f v = *(const v4f*)(h + (size_t)s * DH + lane * 4);
    float* p = aggr + (size_t)d * DH + lane * 4;
    atomAddF(p + 0, v.x);
    atomAddF(p + 1, v.y);
    atomAddF(p + 2, v.z);
    atomAddF(p + 3, v.w);
}

// ---------------- fused GIN layer ----------------
// per 16-row tile: z = relu(BN(A @ W1 + b1)); h = relu(z @ W2 + b2);
// h -> hout, h atomically accumulated into pooled[batch[row]].
// 8 waves per block, wave w owns output columns [16w, 16w+16).
// f32 WMMA 16x16x4: K-loop of 32 steps per 128-deep GEMM.
// A tile staged via GLOBAL_LOAD_ASYNC_TO_LDS_B128 (ASYNCcnt) when available.
__global__ __launch_bounds__(256)
void gin_layer_kernel(const float* __restrict__ aggr,
                      const float* __restrict__ W1, const float* __restrict__ b1,
                      const float* __restrict__ gamma, const float* __restrict__ beta,
                      const float* __restrict__ rmean, const float* __restrict__ rvar,
                      const float* __restrict__ W2, const float* __restrict__ b2,
                      const int* __restrict__ batch,
                      float* __restrict__ hout, float* __restrict__ pooled,
                      int layerOff, int nTiles) {
    __shared__ float As[TM][PADH];
    __shared__ float Zs[TM][PADH];

    const int tid   = threadIdx.x;
    const int wave  = tid >> 5;
    const int lane  = tid & 31;
    const int n16   = lane & 15;
    const int khalf = (lane >> 4) << 1;        // 0 for lanes 0-15, 2 for lanes 16-31
    const int rbase = (lane >> 4) << 3;        // C/D rows: r (lo lanes) / r+8 (hi lanes)
    const int col   = wave * 16 + n16;         // this lane's output column

    // ---- preload B fragments for both GEMMs into registers (loop-invariant) ----
    v2f B1[32], B2[32];
#pragma unroll
    for (int k = 0; k < 32; ++k) {
        const int kr = 4 * k + khalf;
        B1[k].x = W1[(size_t)kr * DH + col];
        B1[k].y = W1[(size_t)(kr + 1) * DH + col];
        B2[k].x = W2[(size_t)kr * DH + col];
        B2[k].y = W2[(size_t)(kr + 1) * DH + col];
    }
    // folded BatchNorm (eval): z = sc * (A@W1)[col] + sh
    const float sc    = gamma[col] * __frsqrt_rn(rvar[col] + 1e-5f);
    const float sh    = (b1[col] - rmean[col]) * sc + beta[col];
    const float bias2 = b2[col];

    for (int tile = blockIdx.x; tile < nTiles; tile += gridDim.x) {
        __syncthreads();   // previous iteration done reading As/Zs
        {
            const float* nxt = aggr + (size_t)(tile + gridDim.x) * TM * DH;
            __builtin_prefetch(nxt, 0, 0);   // global_prefetch_b8 for next tile
            // cooperative staging of 16x128 A tile (each thread: 8 floats = 2 x b128)
            const int r    = tid >> 4;
            const int cpos = (tid & 15) * 8;
            const float* s = aggr + (size_t)tile * TM * DH + r * DH + cpos;
#if HAVE_ASYNC_TO_LDS
            __builtin_amdgcn_global_load_async_to_lds_b128(
                (gas_v4i*)(s),     (las_v4i*)&As[r][cpos],     0, 0);
            __builtin_amdgcn_global_load_async_to_lds_b128(
                (gas_v4i*)(s + 4), (las_v4i*)&As[r][cpos + 4], 0, 0);
            waitAsync0();
#else
            *(v4f*)&As[r][cpos]     = *(const v4f*)(s);
            *(v4f*)&As[r][cpos + 4] = *(const v4f*)(s + 4);
#endif
        }
        __syncthreads();   // As ready (all waves' async transfers complete)

        // ---- GEMM1: acc = A(16x128) @ W1cols(128x16) ----
        v8f acc = {};
#pragma unroll
        for (int k = 0; k < 32; ++k) {
            const v2f a = *(const v2f*)&As[n16][4 * k + khalf];
            acc = __builtin_amdgcn_wmma_f32_16x16x4_f32(false, a, false, B1[k],
                                                        (short)0, acc, false, false);
        }
        // BN + ReLU -> stage z tile
#pragma unroll
        for (int r = 0; r < 8; ++r) {
            float z = acc[r] * sc + sh;
            Zs[r + rbase][col] = z > 0.0f ? z : 0.0f;
        }
        __syncthreads();   // Zs ready

        // ---- GEMM2: acc2 = Z(16x128) @ W2cols(128x16) ----
        v8f acc2 = {};
#pragma unroll
        for (int k = 0; k < 32; ++k) {
            const v2f a = *(const v2f*)&Zs[n16][4 * k + khalf];
            acc2 = __builtin_amdgcn_wmma_f32_16x16x4_f32(false, a, false, B2[k],
                                                         (short)0, acc2, false, false);
        }
        // epilogue: ReLU, store h, pooled atomic add
        const int row0 = tile * TM;
#pragma unroll
        for (int r = 0; r < 8; ++r) {
            float hv = acc2[r] + bias2;
            hv = hv > 0.0f ? hv : 0.0f;
            const int row = row0 + r + rbase;
            hout[(size_t)row * DH + col] = hv;
            const int g = batch[row];
            atomAddF(&pooled[(size_t)g * DCAT + layerOff + col], hv);
        }
    }
}

// ---------------- head: hfin = relu(pooled @ Wfin + bfin) ----------------
__global__ __launch_bounds__(DCAT)
void gin_fin_kernel(const float* __restrict__ pooled, const float* __restrict__ Wfin,
                    const float* __restrict__ bfin, float* __restrict__ hfin) {
    __shared__ float rowS[DCAT];
    const int g = blockIdx.x;
    const int t = threadIdx.x;
    rowS[t] = pooled[(size_t)g * DCAT + t];
    __syncthreads();
    float acc = bfin[t];
#pragma unroll 4
    for (int k = 0; k < DCAT; ++k)
        acc = fmaf(rowS[k], Wfin[(size_t)k * DCAT + t], acc);
    hfin[(size_t)g * DCAT + t] = fmaxf(acc, 0.0f);
}

// ---------------- head: logits + log_softmax ----------------
__global__ __launch_bounds__(32)
void gin_out_kernel(const float* __restrict__ hfin, const float* __restrict__ Wout,
                    const float* __restrict__ bout, float* __restrict__ out) {
    __shared__ float rowS[DCAT];
    __shared__ float lg[NC];
    const int g = blockIdx.x;
    const int t = threadIdx.x;
    for (int k = t; k < DCAT; k += 32) rowS[k] = hfin[(size_t)g * DCAT + k];
    __syncthreads();
    if (t < NC) {
        float acc = bout[t];
        for (int k = 0; k < DCAT; ++k)
            acc = fmaf(rowS[k], Wout[(size_t)k * NC + t], acc);
        lg[t] = acc;
        out[(size_t)g * NC + t] = acc;               // logits
    }
    __syncthreads();
    if (t < NC) {
        float m = lg[0];
#pragma unroll
        for (int j = 1; j < NC; ++j) m = fmaxf(m, lg[j]);
        float s = 0.0f;
#pragma unroll
        for (int j = 0; j < NC; ++j) s += __expf(lg[j] - m);
        out[(size_t)NG * NC + (size_t)g * NC + t] = lg[t] - m - __logf(s);  // log_softmax
    }
}

// ---------------- launcher ----------------
extern "C" void kernel_launch(void* const* d_in, const int* in_sizes, int n_in,
                              void* d_out, int out_size, void* d_ws, size_t ws_size,
                              hipStream_t stream) {
    const float* x     = (const float*)d_in[0];
    const int*   ei    = (const int*)d_in[1];
    const int*   batch = (const int*)d_in[2];
    const float* W1    = (const float*)d_in[4];
    const float* b1    = (const float*)d_in[5];
    const float* gamma = (const float*)d_in[6];
    const float* beta  = (const float*)d_in[7];
    const float* rmean = (const float*)d_in[8];
    const float* rvar  = (const float*)d_in[9];
    const float* W2    = (const float*)d_in[10];
    const float* b2    = (const float*)d_in[11];
    const float* Wfin  = (const float*)d_in[12];
    const float* bfin  = (const float*)d_in[13];
    const float* Wout  = (const float*)d_in[14];
    const float* bout  = (const float*)d_in[15];
    float* out = (float*)d_out;

    const int nNodes = in_sizes[0] / DH;   // 100000
    const int nEdges = in_sizes[1] / 2;    // 3200000
    const int* srcI = ei;
    const int* dstI = ei + nEdges;

    // workspace layout (floats)
    float* aggr   = (float*)d_ws;                       // nNodes*DH
    float* hA     = aggr + (size_t)nNodes * DH;         // nNodes*DH
    float* hB     = hA   + (size_t)nNodes * DH;         // nNodes*DH
    float* pooled = hB   + (size_t)nNodes * DH;         // NG*DCAT
    float* hfin   = pooled + (size_t)NG * DCAT;         // NG*DCAT

    gin_zero_kernel<<<(NG * DCAT + 255) / 256, 256, 0, stream>>>(pooled, NG * DCAT);

    const int nTiles = nNodes / TM;
    const int n4     = nNodes * DH / 4;
    const long long edgeThreads = (long long)nEdges * 32;
    const int edgeBlocks = (int)((edgeThreads + 255) / 256);

    const float* hcur = x;
    float* bufs[2] = { hA, hB };
    for (int i = 0; i < NL; ++i) {
        // aggr = h (self term), then += neighbor sum
        gin_copy4_kernel<<<(n4 + 255) / 256, 256, 0, stream>>>(aggr, hcur, n4);
        gin_edge_kernel<<<edgeBlocks, 256, 0, stream>>>(hcur, srcI, dstI, aggr, nEdges);
        float* hnext = bufs[i & 1];
        gin_layer_kernel<<<512, 256, 0, stream>>>(
            aggr, W1 + (size_t)i * DH * DH, b1 + i * DH,
            gamma + i * DH, beta + i * DH, rmean + i * DH, rvar + i * DH,
            W2 + (size_t)i * DH * DH, b2 + i * DH, batch,
            hnext, pooled, i * DH, nTiles);
        hcur = hnext;
    }
    gin_fin_kernel<<<NG, DCAT, 0, stream>>>(pooled, Wfin, bfin, hfin);
    gin_out_kernel<<<NG, 32, 0, stream>>>(hfin, Wout, bout, out);
}